// StatisticalHead_50852412785061
// MI455X (gfx1250) — compile-verified
//
#include <hip/hip_runtime.h>
#include <math.h>

typedef __attribute__((ext_vector_type(2))) float v2f;
typedef __attribute__((ext_vector_type(8))) float v8f;

#define SDIM 256
#define FDIM 64
#define HDIM 64
#define IDIM 65

__device__ __forceinline__ float relu_f(float x){ return fmaxf(x, 0.0f); }

// ws layout (floats):
// [0:64) kA  [64:128) kU  [128:192) kC  [192:256) vA  [256:320) vU  [320:384) vC
// [384:2432) gctx  [2432:133504) qk  [133504:135552) attbias   (~530 KB total)

__global__ void fuse_kernel(const float* __restrict__ kW1, const float* __restrict__ kb1,
                            const float* __restrict__ vW1, const float* __restrict__ vb1,
                            const float* __restrict__ tW,  const float* __restrict__ tb,
                            float* __restrict__ ws) {
  int h = threadIdx.x; // 64 threads
  const float* krow = kW1 + h * IDIM;
  const float* vrow = vW1 + h * IDIM;
  float ku = 0.f, kc = 0.f, vu = 0.f, vc = 0.f;
  for (int j = 0; j < HDIM; ++j) {
    float kw = krow[1 + j], vw = vrow[1 + j];
    ku = fmaf(kw, tW[j], ku);
    kc = fmaf(kw, tb[j], kc);
    vu = fmaf(vw, tW[j], vu);
    vc = fmaf(vw, tb[j], vc);
  }
  ws[h]        = krow[0];
  ws[64 + h]   = ku;
  ws[128 + h]  = kc + kb1[h];
  ws[192 + h]  = vrow[0];
  ws[256 + h]  = vu;
  ws[320 + h]  = vc + vb1[h];
}

__global__ void gctx_kernel(const float* __restrict__ values, const float* __restrict__ masks,
                            float* __restrict__ gctx) {
  int pair = blockIdx.x * blockDim.x + threadIdx.x; // 2048
  int b = pair >> 6, f = pair & 63;
  const float* vb = values + (size_t)b * SDIM * FDIM + f;
  const float* mb = masks  + (size_t)b * SDIM * FDIM + f;
  float s = 0.f, c = 0.f;
  for (int t = 0; t < SDIM; ++t) {
    float m = mb[t * FDIM];
    s = fmaf(vb[t * FDIM], m, s);
    c += m;
  }
  gctx[pair] = s / fmaxf(c, 1.0f);
}

// Fused query/qk GEMMs via V_WMMA_F32_16X16X4_F32.
// Each wave handles 16 rows (m0..m0+15) of the 2048-row problem.
__global__ __launch_bounds__(128) void qk_kernel(
    const float* __restrict__ gctx, const float* __restrict__ qW1, const float* __restrict__ qb1,
    const float* __restrict__ qW2, const float* __restrict__ qb2,
    const float* __restrict__ kW2, const float* __restrict__ kb2,
    float* __restrict__ qk, float* __restrict__ attbias) {
  __shared__ float ldsQ[4][16][68]; // padded stride 68 to avoid bank conflicts
  int lane = threadIdx.x & 31;
  int wv   = threadIdx.x >> 5;
  int m0   = (blockIdx.x * 4 + wv) * 16;
  int lm   = lane & 15;
  int hi8  = (lane >= 16) ? 8 : 0;
  int koff = (lane >= 16) ? 2 : 0;

  // A fragments of X = q_h1 = relu(gctx*qW1[:,0] + qb1), rows m0..m0+15, K=0..63
  // f32 16x4 A layout: lanes 0-15 hold (K=kb,kb+1), lanes 16-31 hold (K=kb+2,kb+3)
  float g = gctx[m0 + lm];
  v2f a[16];
#pragma unroll
  for (int k4 = 0; k4 < 16; ++k4) {
    int h = k4 * 4 + koff;
    a[k4].x = relu_f(fmaf(g, qW1[h * IDIM], qb1[h]));
    a[k4].y = relu_f(fmaf(g, qW1[(h + 1) * IDIM], qb1[h + 1]));
  }

  // Phase 1: Q = X @ qW2^T + qb2  -> LDS (D layout: VGPR g -> row g / g+8)
  for (int nt = 0; nt < 4; ++nt) {
    int n = nt * 16 + lm;
    float bias = qb2[n];
    v8f c;
#pragma unroll
    for (int i = 0; i < 8; ++i) c[i] = bias;
#pragma unroll
    for (int k4 = 0; k4 < 16; ++k4) {
      // B[k,n] = qW2[n,k]: consecutive k -> contiguous in memory
      const float* p = qW2 + n * HDIM + k4 * 4 + koff;
      v2f bf; bf.x = p[0]; bf.y = p[1];
      c = __builtin_amdgcn_wmma_f32_16x16x4_f32(false, a[k4], false, bf,
                                                (short)0, c, false, false);
    }
#pragma unroll
    for (int gi = 0; gi < 8; ++gi)
      ldsQ[wv][gi + hi8][nt * 16 + lm] = c[gi];
  }
  __syncthreads();

  // attbias[m] = Q[m,:] . kb2
  if (lane < 16) {
    float s = 0.f;
    for (int h = 0; h < HDIM; ++h) s = fmaf(ldsQ[wv][lane][h], kb2[h], s);
    attbias[m0 + lane] = s;
  }

  // Phase 2: QK = Q @ kW2  (B[k,n] = kW2[k,n])
  for (int nt = 0; nt < 4; ++nt) {
    int n = nt * 16 + lm;
    v8f c;
#pragma unroll
    for (int i = 0; i < 8; ++i) c[i] = 0.0f;
#pragma unroll
    for (int k4 = 0; k4 < 16; ++k4) {
      int kk = k4 * 4 + koff;
      v2f af; af.x = ldsQ[wv][lm][kk]; af.y = ldsQ[wv][lm][kk + 1];
      v2f bf; bf.x = kW2[kk * HDIM + n]; bf.y = kW2[(kk + 1) * HDIM + n];
      c = __builtin_amdgcn_wmma_f32_16x16x4_f32(false, af, false, bf,
                                                (short)0, c, false, false);
    }
#pragma unroll
    for (int gi = 0; gi < 8; ++gi)
      qk[(size_t)(m0 + hi8 + gi) * HDIM + n] = c[gi];
  }
}

// One wave32 per (b,f). Lanes own 8 s-values each (s = c*32 + lane).
__global__ __launch_bounds__(256) void attn_kernel(
    const float* __restrict__ values, const float* __restrict__ times,
    const float* __restrict__ masks,  const float* __restrict__ ws,
    const float* __restrict__ qk,     const float* __restrict__ attbias,
    const float* __restrict__ vW2,    const float* __restrict__ vb2,
    float* __restrict__ out) {
  int lane = threadIdx.x & 31;
  int pair = blockIdx.x * 8 + (threadIdx.x >> 5);
  int b = pair >> 6, f = pair & 63;
  const float NEG_INF = -__builtin_inff();

  float kA0 = ws[lane],       kA1 = ws[lane + 32];
  float kU0 = ws[64 + lane],  kU1 = ws[96 + lane];
  float kC0 = ws[128 + lane], kC1 = ws[160 + lane];
  float vA0 = ws[192 + lane], vA1 = ws[224 + lane];
  float vU0 = ws[256 + lane], vU1 = ws[288 + lane];
  float vC0 = ws[320 + lane], vC1 = ws[352 + lane];
  float qk0 = qk[(size_t)pair * 64 + lane];
  float qk1 = qk[(size_t)pair * 64 + 32 + lane];
  float bias = attbias[pair];

  float vv[8], tt[8], mm[8];
#pragma unroll
  for (int c = 0; c < 8; ++c) {
    int s = c * 32 + lane;
    size_t idx = ((size_t)b * SDIM + s) * FDIM + f;
    float m = masks[idx];
    mm[c] = m;
    vv[c] = values[idx] * m;
    tt[c] = times[b * SDIM + s];
  }

  // Pass 1: att[s] = (relu(v*kA + t*kU + kC) . qk + query.kb2) / 8
  float dv[8];
#pragma unroll
  for (int c = 0; c < 8; ++c) dv[c] = 0.f;
  for (int h = 0; h < 64; ++h) {
    float ka = __shfl((h < 32) ? kA0 : kA1, h & 31, 32);
    float ku = __shfl((h < 32) ? kU0 : kU1, h & 31, 32);
    float kc = __shfl((h < 32) ? kC0 : kC1, h & 31, 32);
    float qh = __shfl((h < 32) ? qk0 : qk1, h & 31, 32);
#pragma unroll
    for (int c = 0; c < 8; ++c) {
      float x = fmaf(vv[c], ka, fmaf(tt[c], ku, kc));
      x = fmaxf(x, 0.f);
      dv[c] = fmaf(x, qh, dv[c]);
    }
  }
  float att[8];
  float cnt = 0.f;
#pragma unroll
  for (int c = 0; c < 8; ++c) {
    cnt += mm[c];
    att[c] = (mm[c] != 0.f) ? (dv[c] + bias) * 0.125f : NEG_INF;
  }
  float mx = att[0];
#pragma unroll
  for (int c = 1; c < 8; ++c) mx = fmaxf(mx, att[c]);
#pragma unroll
  for (int off = 16; off > 0; off >>= 1) {
    mx = fmaxf(mx, __shfl_xor(mx, off, 32));
    cnt += __shfl_xor(cnt, off, 32);
  }
  float wgt[8];
  float se = 0.f;
#pragma unroll
  for (int c = 0; c < 8; ++c) {
    float e = (mm[c] != 0.f) ? __expf(att[c] - mx) : 0.f;
    wgt[c] = e;
    se += e;
  }
#pragma unroll
  for (int off = 16; off > 0; off >>= 1) se += __shfl_xor(se, off, 32);
  float inv = (cnt > 0.f) ? (1.f / se) : 0.f; // all-masked row -> w = 0
#pragma unroll
  for (int c = 0; c < 8; ++c) wgt[c] *= inv;

  // Pass 2: acc[h] = sum_s w[s] * relu(v*vA + t*vU + vC)[h]
  float acc0 = 0.f, acc1 = 0.f;
  for (int h = 0; h < 64; ++h) {
    float va = __shfl((h < 32) ? vA0 : vA1, h & 31, 32);
    float vu = __shfl((h < 32) ? vU0 : vU1, h & 31, 32);
    float vc = __shfl((h < 32) ? vC0 : vC1, h & 31, 32);
    float ph = 0.f;
#pragma unroll
    for (int c = 0; c < 8; ++c) {
      float y = fmaf(vv[c], va, fmaf(tt[c], vu, vc));
      y = fmaxf(y, 0.f);
      ph = fmaf(wgt[c], y, ph);
    }
#pragma unroll
    for (int off = 16; off > 0; off >>= 1) ph += __shfl_xor(ph, off, 32);
    if ((h & 31) == lane) { if (h < 32) acc0 = ph; else acc1 = ph; }
  }

  // Epilogue: out[o] = acc . vW2[o,:] + (cnt>0)*vb2[o]
  int o0 = lane, o1 = lane + 32;
  float s0 = 0.f, s1 = 0.f;
  for (int h = 0; h < 64; ++h) {
    float aL = __shfl((h < 32) ? acc0 : acc1, h & 31, 32);
    s0 = fmaf(aL, vW2[o0 * 64 + h], s0);
    s1 = fmaf(aL, vW2[o1 * 64 + h], s1);
  }
  float sw = (cnt > 0.f) ? 1.f : 0.f;
  out[(size_t)pair * 64 + o0] = s0 + sw * vb2[o0];
  out[(size_t)pair * 64 + o1] = s1 + sw * vb2[o1];
}

extern "C" void kernel_launch(void* const* d_in, const int* in_sizes, int n_in,
                              void* d_out, int out_size, void* d_ws, size_t ws_size,
                              hipStream_t stream) {
  const float* values = (const float*)d_in[0];
  const float* times  = (const float*)d_in[1];
  const float* masks  = (const float*)d_in[2];
  const float* qW1 = (const float*)d_in[3];
  const float* qb1 = (const float*)d_in[4];
  const float* qW2 = (const float*)d_in[5];
  const float* qb2 = (const float*)d_in[6];
  const float* kW1 = (const float*)d_in[7];
  const float* kb1 = (const float*)d_in[8];
  const float* kW2 = (const float*)d_in[9];
  const float* kb2 = (const float*)d_in[10];
  const float* vW1 = (const float*)d_in[11];
  const float* vb1 = (const float*)d_in[12];
  const float* vW2 = (const float*)d_in[13];
  const float* vb2 = (const float*)d_in[14];
  const float* tW  = (const float*)d_in[15];
  const float* tb  = (const float*)d_in[16];
  float* out = (float*)d_out;

  float* ws      = (float*)d_ws;
  float* gctx    = ws + 384;
  float* qkbuf   = ws + 2432;
  float* attbias = ws + 2432 + 2048 * 64;

  fuse_kernel<<<1, 64, 0, stream>>>(kW1, kb1, vW1, vb1, tW, tb, ws);
  gctx_kernel<<<8, 256, 0, stream>>>(values, masks, gctx);
  qk_kernel<<<32, 128, 0, stream>>>(gctx, qW1, qb1, qW2, qb2, kW2, kb2, qkbuf, attbias);
  attn_kernel<<<256, 256, 0, stream>>>(values, times, masks, ws, qkbuf, attbias, vW2, vb2, out);
}